// SymRegLayer1_39926015983921
// MI455X (gfx1250) — compile-verified
//
#include <hip/hip_runtime.h>
#include <hip/hip_bf16.h>
#include <stdint.h>

typedef __attribute__((ext_vector_type(2))) float v2f;
typedef __attribute__((ext_vector_type(8))) float v8f;

#define DIN  128
#define DOUT 64
#define DCAT 192

// ---------------------------------------------------------------------------
// GEMM1: h[N,64] = x[N,128] @ W_lin[128,64]   (WMMA f32 16x16x4)
// Block = 128 threads = 4 waves; each wave owns one 16x16 output tile
// (4 tiles cover DOUT=64). One block covers 16 rows. N=50000 = 3125*16.
// ---------------------------------------------------------------------------
__global__ void __launch_bounds__(128) gemm_lin1(const float* __restrict__ x,
                                                 const float* __restrict__ Wlin,
                                                 float* __restrict__ h, int Nn) {
    __shared__ float Bs[DIN * DOUT];           // 32 KB
    int tid = threadIdx.x;
    for (int i = tid; i < DIN * DOUT; i += 128) Bs[i] = Wlin[i];
    __syncthreads();

    int wave = tid >> 5;
    int lane = tid & 31;
    int row0 = blockIdx.x << 4;
    int col0 = wave << 4;
    int mn   = lane & 15;          // M for A-frag, N for B/D-frag
    int koff = (lane >> 4) << 1;   // lanes 16-31 hold K+2,K+3

    int arow = row0 + mn; if (arow >= Nn) arow = Nn - 1;   // clamp tail reads
    const float* ap = x + (size_t)arow * DIN;

    v8f acc = {};
    for (int k = 0; k < DIN; k += 4) {
        v2f a, b;
        a.x = ap[k + koff];
        a.y = ap[k + koff + 1];
        b.x = Bs[(k + koff)     * DOUT + col0 + mn];
        b.y = Bs[(k + koff + 1) * DOUT + col0 + mn];
        acc = __builtin_amdgcn_wmma_f32_16x16x4_f32(false, a, false, b,
                                                    (short)0, acc, false, false);
    }

    int mbase = (lane >> 4) << 3;  // D: vgpr r -> M = mbase + r
#pragma unroll
    for (int r = 0; r < 8; ++r) {
        int row = row0 + mbase + r;
        if (row < Nn) h[(size_t)row * DOUT + col0 + mn] = acc[r];
    }
}

// ---------------------------------------------------------------------------
// GEMM2: out[N,64] = aggcat[N,192] @ conv_w^T[192,64] + bias_eff
// conv_w is [64,192] row-major; stage transposed into LDS (48 KB).
// C initialized from bias_eff (per-output-column constant).
// ---------------------------------------------------------------------------
__global__ void __launch_bounds__(128) gemm_conv1d(const float* __restrict__ aggcat,
                                                   const float* __restrict__ convw,
                                                   const float* __restrict__ biaseff,
                                                   float* __restrict__ out, int Nn) {
    __shared__ float Bs[DCAT * DOUT];          // 48 KB, Bs[k*64+n] = convw[n*192+k]
    int tid = threadIdx.x;
    for (int j = tid; j < DOUT * DCAT; j += 128) {
        int n = j / DCAT, k = j % DCAT;
        Bs[k * DOUT + n] = convw[j];
    }
    __syncthreads();

    int wave = tid >> 5;
    int lane = tid & 31;
    int row0 = blockIdx.x << 4;
    int col0 = wave << 4;
    int mn   = lane & 15;
    int koff = (lane >> 4) << 1;

    int arow = row0 + mn; if (arow >= Nn) arow = Nn - 1;
    const float* ap = aggcat + (size_t)arow * DCAT;

    float be = biaseff[col0 + mn];
    v8f acc = {be, be, be, be, be, be, be, be};
    for (int k = 0; k < DCAT; k += 4) {
        v2f a, b;
        a.x = ap[k + koff];
        a.y = ap[k + koff + 1];
        b.x = Bs[(k + koff)     * DOUT + col0 + mn];
        b.y = Bs[(k + koff + 1) * DOUT + col0 + mn];
        acc = __builtin_amdgcn_wmma_f32_16x16x4_f32(false, a, false, b,
                                                    (short)0, acc, false, false);
    }

    int mbase = (lane >> 4) << 3;
#pragma unroll
    for (int r = 0; r < 8; ++r) {
        int row = row0 + mbase + r;
        if (row < Nn) out[(size_t)row * DOUT + col0 + mn] = acc[r];
    }
}

// ---------------------------------------------------------------------------
// Helpers
// ---------------------------------------------------------------------------
__global__ void zero_f32(float* p, int n) {
    int i = blockIdx.x * 256 + threadIdx.x;
    if (i < n) p[i] = 0.0f;
}

// deg[col] += w (or 1.0)
__global__ void deg_acc(const int* __restrict__ ei, const float* __restrict__ w,
                        float* __restrict__ deg, int Ee) {
    int e = blockIdx.x * 256 + threadIdx.x;
    if (e >= Ee) return;
    int col = ei[Ee + e];
    atomicAdd(&deg[col], w ? w[e] : 1.0f);
}

// dinv = (deg + 1)^-0.5  (self-loop weight 1 folded in; deg+1 > 0 always)
__global__ void dinv_k(const float* __restrict__ deg, float* __restrict__ dinv, int n) {
    int i = blockIdx.x * 256 + threadIdx.x;
    if (i >= n) return;
    float d = deg[i] + 1.0f;
    dinv[i] = (d > 0.0f) ? rsqrtf(d) : 0.0f;
}

// Initialize aggcat with the self-loop contribution dinv_c[i]^2 * h[i]
__global__ void init_aggcat(const float* __restrict__ h, const float* __restrict__ dinv,
                            float* __restrict__ aggcat, int Nn) {
    int i = blockIdx.x * 256 + threadIdx.x;
    if (i >= Nn * DOUT) return;
    int node = i >> 6, f = i & 63;
    float hv = h[i];
#pragma unroll
    for (int c = 0; c < 3; ++c) {
        float dv = dinv[c * Nn + node];
        aggcat[(size_t)node * DCAT + c * DOUT + f] = dv * dv * hv;
    }
}

// Per-edge message: aggseg[col,:] += dinv[row]*w*dinv[col] * h[row,:]
// 64 threads per edge (one per feature).
__global__ void edge_scatter(const int* __restrict__ ei, const float* __restrict__ w,
                             const float* __restrict__ dinv, const float* __restrict__ h,
                             float* __restrict__ aggseg, int Ee) {
    long long t = (long long)blockIdx.x * 256 + threadIdx.x;
    int e = (int)(t >> 6);
    if (e >= Ee) return;
    int f   = (int)(t & 63);
    int row = ei[e];
    int col = ei[Ee + e];
    float ww = w ? w[e] : 1.0f;
    float nm = dinv[row] * ww * dinv[col];
    atomicAdd(&aggseg[(size_t)col * DCAT + f], nm * h[(size_t)row * DOUT + f]);
}

// bias_eff[n] = conv_b[n] + sum_k bias1[k%64] * conv_w[n*192+k]
__global__ void bias_eff_k(const float* __restrict__ bias1, const float* __restrict__ convw,
                           const float* __restrict__ convb, float* __restrict__ biaseff) {
    int n = threadIdx.x;
    if (n >= DOUT) return;
    float s = convb[n];
    for (int k = 0; k < DCAT; ++k) s += bias1[k & 63] * convw[n * DCAT + k];
    biaseff[n] = s;
}

// ---------------------------------------------------------------------------
extern "C" void kernel_launch(void* const* d_in, const int* in_sizes, int n_in,
                              void* d_out, int out_size, void* d_ws, size_t ws_size,
                              hipStream_t stream) {
    const float* x          = (const float*)d_in[0];
    const int*   edge_index = (const int*)  d_in[1];
    const int*   edge_in    = (const int*)  d_in[2];
    const float* in_w       = (const float*)d_in[3];
    const int*   edge_out   = (const int*)  d_in[4];
    const float* out_w      = (const float*)d_in[5];
    const float* W_lin      = (const float*)d_in[6];
    const float* bias1      = (const float*)d_in[7];
    const float* conv_w     = (const float*)d_in[8];
    const float* conv_b     = (const float*)d_in[9];

    int Nn = in_sizes[0] / DIN;      // 50000
    int Ee = in_sizes[3];            // 800000
    float* out = (float*)d_out;

    // workspace layout
    float* ws      = (float*)d_ws;
    float* h       = ws;                              // N*64
    float* aggcat  = h + (size_t)Nn * DOUT;           // N*192
    float* deg     = aggcat + (size_t)Nn * DCAT;      // 3*N
    float* dinv    = deg + 3 * (size_t)Nn;            // 3*N
    float* biaseff = dinv + 3 * (size_t)Nn;           // 64

    int rowBlocks = (Nn + 15) / 16;

    // 1) h = x @ W_lin  (WMMA)
    gemm_lin1<<<rowBlocks, 128, 0, stream>>>(x, W_lin, h, Nn);

    // 2) degrees
    int zn = 3 * Nn;
    zero_f32<<<(zn + 255) / 256, 256, 0, stream>>>(deg, zn);
    int egrid = (Ee + 255) / 256;
    deg_acc<<<egrid, 256, 0, stream>>>(edge_index, nullptr, deg,          Ee);
    deg_acc<<<egrid, 256, 0, stream>>>(edge_in,    in_w,    deg + Nn,     Ee);
    deg_acc<<<egrid, 256, 0, stream>>>(edge_out,   out_w,   deg + 2 * Nn, Ee);
    dinv_k<<<(zn + 255) / 256, 256, 0, stream>>>(deg, dinv, zn);

    // 3) self-loop init of concat buffer
    init_aggcat<<<(Nn * DOUT + 255) / 256, 256, 0, stream>>>(h, dinv, aggcat, Nn);

    // 4) edge aggregation (64 threads per edge)
    long long tot = (long long)Ee * DOUT;
    int sgrid = (int)((tot + 255) / 256);
    edge_scatter<<<sgrid, 256, 0, stream>>>(edge_index, nullptr, dinv,          h, aggcat,            Ee);
    edge_scatter<<<sgrid, 256, 0, stream>>>(edge_in,    in_w,    dinv + Nn,     h, aggcat + DOUT,     Ee);
    edge_scatter<<<sgrid, 256, 0, stream>>>(edge_out,   out_w,   dinv + 2 * Nn, h, aggcat + 2 * DOUT, Ee);

    // 5) folded bias, 6) final GEMM (WMMA) -> d_out
    bias_eff_k<<<1, 64, 0, stream>>>(bias1, conv_w, conv_b, biaseff);
    gemm_conv1d<<<rowBlocks, 128, 0, stream>>>(aggcat, conv_w, biaseff, out, Nn);
}